// SupConLoss_53798760350423
// MI455X (gfx1250) — compile-verified
//
#include <hip/hip_runtime.h>
#include <math.h>

// SupCon loss for MI455X (gfx1250, wave32).
//
// feat[n][c] = features[b][c][h][w], n = b*1024 + h*32 + w  (B=4,C=128,H=W=32, N=4096)
// logits = feat @ feat^T / 0.07  -> fused per-16x16-tile epilogue, never materialized.
//
// GEMM runs on V_WMMA_F32_16X16X4_F32 (exact f32, matches reference math):
//  A 16x4 layout (ISA 7.12.2): lane = 16*hi + lo holds {K=2hi, K=2hi+1} of row lo.
//  B  4x16 layout: element v of the pair holds K = v + 2hi, col = lo  -> for a Gram
//     matrix B loads are the same per-lane pattern as A with the column-tile base.
//  C/D 16x16: acc[v] = logits[row0 + v + 8*hi][j0 + lo].

typedef __attribute__((ext_vector_type(2))) float v2f;
typedef __attribute__((ext_vector_type(8))) float v8f;

#define TEMP_INV  (1.0f / 0.07f)
#define EPSF      1e-6f
#define N_TOT     4096
#define KSTEPS    32          // 128 / 4

// labels int64 [2][2][32][32]; lab[n] follows cat(unbind(dim=1)) ordering.
__device__ __forceinline__ int lab_of(const long long* __restrict__ L, int n) {
    int m   = (n >= 2048) ? (n - 2048) : n;
    int idx = ((m >> 10) << 11) + ((n >= 2048) ? 1024 : 0) + (m & 1023);
    return (int)L[idx];
}

// feat(n, c) in the original [B][C][H][W] layout (131072 = C*H*W = 1<<17).
__device__ __forceinline__ float feat_at(const float* __restrict__ F, int base, int c) {
    return F[base + (c << 10)];
}
__device__ __forceinline__ int feat_base(int n) {
    return ((n >> 10) << 17) + (n & 1023);
}

__global__ __launch_bounds__(256) void supcon_main(const float* __restrict__ F,
                                                   const long long* __restrict__ L,
                                                   float* __restrict__ partial) {
    __shared__ float rowS[8][16];   // per-wave row accumulators (fixed-order reduce)
    __shared__ float rowP[8][16];
    __shared__ float rowM[8][16];
    __shared__ int   clsI[4];
    __shared__ float clsF[4];
    __shared__ float mlpp[16];

    const int tid  = threadIdx.x;
    const int wave = tid >> 5;
    const int lane = tid & 31;
    const int lo   = lane & 15;
    const int hi   = lane >> 4;
    const int row0 = blockIdx.x * 16;

    // ---- per-block class histogram (integer atomics: deterministic) ----
    if (tid < 4) clsI[tid] = 0;
    __syncthreads();
    for (int i = tid; i < N_TOT; i += 256) atomicAdd(&clsI[lab_of(L, i)], 1);
    __syncthreads();
    if (tid < 4) clsF[tid] = (float)clsI[tid];
    __syncthreads();

    // ---- preload this wave's A tile (16 rows x 128 K) into registers ----
    const int abase = feat_base(row0 + lo);
    v2f A[KSTEPS];
#pragma unroll
    for (int t = 0; t < KSTEPS; ++t) {
        const int k = 4 * t + 2 * hi;
        A[t].x = feat_at(F, abase, k);
        A[t].y = feat_at(F, abase, k + 1);
    }

    // labels of the 8 rows this lane's accumulator elements belong to
    int labr[8];
#pragma unroll
    for (int v = 0; v < 8; ++v) labr[v] = lab_of(L, row0 + v + 8 * hi);

    float S[8], P[8], Mc[8];
#pragma unroll
    for (int v = 0; v < 8; ++v) { S[v] = 0.f; P[v] = 0.f; Mc[v] = 0.f; }

    // ---- sweep column tiles: WMMA chain over K, fused epilogue ----
    for (int ct = wave; ct < 256; ct += 8) {
        const int   j     = ct * 16 + lo;
        const int   labj  = lab_of(L, j);
        const float cntj  = clsF[labj];
        const int   jbase = feat_base(j);

        v8f acc = {0.f, 0.f, 0.f, 0.f, 0.f, 0.f, 0.f, 0.f};
#pragma unroll
        for (int t = 0; t < KSTEPS; ++t) {
            const int k = 4 * t + 2 * hi;
            v2f b;
            b.x = feat_at(F, jbase, k);
            b.y = feat_at(F, jbase, k + 1);
            acc = __builtin_amdgcn_wmma_f32_16x16x4_f32(
                false, A[t], false, b, (short)0, acc, false, false);
        }

#pragma unroll
        for (int v = 0; v < 8; ++v) {
            const int   r    = row0 + v + 8 * hi;
            const float g    = acc[v] * TEMP_INV;
            const bool  diag = (r == j);
            const bool  same = (labr[v] == labj) && !diag;
            const float w    = cntj - (same ? 1.0f : 0.0f);
            const float e    = expf(g) / w;             // branch-free: EXEC stays full
            S[v]  += diag ? 0.0f : e;
            P[v]  += same ? g    : 0.0f;
            Mc[v] += same ? 1.0f : 0.0f;
        }
    }

    // ---- reduce the 16 column-lanes of each row half (stays within a half) ----
#pragma unroll
    for (int v = 0; v < 8; ++v) {
        float s = S[v], p = P[v], m = Mc[v];
#pragma unroll
        for (int off = 1; off < 16; off <<= 1) {
            s += __shfl_xor(s, off, 32);
            p += __shfl_xor(p, off, 32);
            m += __shfl_xor(m, off, 32);
        }
        if (lo == 0) {
            const int rr = v + 8 * hi;
            rowS[wave][rr] = s;
            rowP[wave][rr] = p;
            rowM[wave][rr] = m;
        }
    }
    __syncthreads();

    // ---- cross-wave reduce in fixed order, per-row log-prob mean ----
    if (tid < 16) {
        float s = 0.f, p = 0.f, m = 0.f;
#pragma unroll
        for (int w = 0; w < 8; ++w) { s += rowS[w][tid]; p += rowP[w][tid]; m += rowM[w][tid]; }
        mlpp[tid] = (p - m * logf(s + EPSF)) / (m + EPSF);
    }
    __syncthreads();
    if (tid == 0) {
        float t = 0.f;
#pragma unroll
        for (int r = 0; r < 16; ++r) t += mlpp[r];
        partial[blockIdx.x] = t;
    }
}

__global__ __launch_bounds__(256) void supcon_finalize(const float* __restrict__ partial,
                                                       float* __restrict__ out) {
    __shared__ float sm[256];
    const int tid = threadIdx.x;
    sm[tid] = partial[tid];
    __syncthreads();
    for (int s = 128; s > 0; s >>= 1) {
        if (tid < s) sm[tid] += sm[tid + s];
        __syncthreads();
    }
    if (tid == 0) out[0] = -sm[0] / 4096.0f;
}

extern "C" void kernel_launch(void* const* d_in, const int* in_sizes, int n_in,
                              void* d_out, int out_size, void* d_ws, size_t ws_size,
                              hipStream_t stream) {
    (void)in_sizes; (void)n_in; (void)out_size; (void)ws_size;
    const float*     F = (const float*)d_in[0];      // features f32 [4,128,32,32]
    const long long* L = (const long long*)d_in[1];  // labels  i64 [2,2,32,32]
    float* partial = (float*)d_ws;                   // 256 floats of scratch

    supcon_main<<<256, 256, 0, stream>>>(F, L, partial);
    supcon_finalize<<<1, 256, 0, stream>>>(partial, (float*)d_out);
}